// Model_38508676775941
// MI455X (gfx1250) — compile-verified
//
#include <hip/hip_runtime.h>
#include <stdint.h>

#define CDIM     5120          // NUM_ROWS * ALPHA_LEN
#define ROWBYTES 20480         // CDIM * 4
#define NEGINF   (-1e30f)
#define DEPTH    8             // TDM pipeline depth (160 KB LDS of 320 KB/WGP)

typedef uint32_t u32x4 __attribute__((ext_vector_type(4)));
typedef uint32_t u32x8 __attribute__((ext_vector_type(8)));

// __shared__ generic pointer -> 32-bit LDS byte offset (low 32 bits of the
// LDS-aperture flat address are the workgroup-relative LDS address).
__device__ __forceinline__ uint32_t lds_byte_off(const void* p) {
  return (uint32_t)(uintptr_t)p;
}

// Issue one Tensor-Data-Mover DMA: a 1-row tile (5120 x fp32 = 20 KB) from
// global memory to LDS. D# built per CDNA5 ISA 8.3-8.6. Executed by ONE wave
// (TDM ignores EXEC; one instruction = one descriptor processed), tracked by
// TENSORcnt (in-order within the wave). Non-temporal: stream has zero reuse.
__device__ __forceinline__ void tdm_load_row(uint32_t lds_off, const float* gptr,
                                             u32x8 g1, u32x4 g23) {
  const uint64_t ga = (uint64_t)(uintptr_t)gptr;
  u32x4 g0;
  g0[0] = 1u;                                            // count=1, user D#
  g0[1] = lds_off;                                       // lds_addr [63:32]
  g0[2] = (uint32_t)ga;                                  // global_addr lo
  g0[3] = ((uint32_t)(ga >> 32) & 0x01FFFFFFu)           // global_addr [56:32]
          | 0x80000000u;                                 // type=2 ("image")
  asm volatile("tensor_load_to_lds %0, %1, %2, %3 th:TH_LOAD_NT"
               :: "s"(g0), "s"(g1), "s"(g23), "s"(g23)
               : "memory");
}

__global__ void zero_out_kernel(float* out) { out[0] = 0.0f; }

// One block per batch element n. 1024 threads = 32 waves.
// Fuses: full-trellis logZ (1024 states, one per thread) + banded CTC logZ
// (nst states), consuming each streamed score row exactly once from LDS.
// Normalisation folds analytically: logZ_ctc_norm = logZ_ctc_raw - logZ_full.
__global__ __launch_bounds__(1024) void crf_loss_fused_kernel(
    const float* __restrict__ scores,   // [T, N, CDIM]
    const int*   __restrict__ targets,  // [N, L]
    const int*   __restrict__ tlen,     // [N]
    float*       __restrict__ out,      // [1]
    int T, int N, int L) {
  __shared__ __align__(16) float sbuf[DEPTH][CDIM];  // 160 KB circular buffer
  __shared__ float abuf[2][1024];                    // trellis alpha ping-pong
  __shared__ float bbuf[2][512];                     // banded alpha ping-pong
  __shared__ float red[1024];                        // reduction scratch
  __shared__ int   tgt_s[512];
  __shared__ int   stay_s[512];
  __shared__ int   move_s[512];

  const int tid = threadIdx.x;
  const int n   = blockIdx.x;
  const int nst = L - 4;                              // band width (396)

  // --- per-batch gather indices for the banded recursion ---
  if (tid < L) {
    int v = targets[n * L + tid] - 1;
    tgt_s[tid] = v < 0 ? 0 : v;
  }
  __syncthreads();
  if (tid < nst) {
    int km = tgt_s[tid] * 256 + tgt_s[tid + 1] * 64 + tgt_s[tid + 2] * 16 +
             tgt_s[tid + 3] * 4 + tgt_s[tid + 4];
    stay_s[tid] = km * 5;
  }
  __syncthreads();
  if (tid >= 1 && tid < nst) move_s[tid] = stay_s[tid] + tgt_s[tid - 1] + 1;

  // --- alpha init ---
  abuf[0][tid] = 0.0f;                                // log-semiring one
  if (tid < 512) bbuf[0][tid] = (tid == 0) ? 0.0f : NEGINF;

  const float*   base = scores + (size_t)n * CDIM;    // row (0, n)
  const uint32_t sb   = lds_byte_off(&sbuf[0][0]);
  const bool     w0   = (tid < 32);                   // wave 0 drives the TDM

  // D# group 1: data_size=4B; tensor 5120 x 1; tile 5120 x 1; stride0 = 5120.
  const u32x8 g1 = {0x00020000u,                      // data_size = 2 (4 bytes)
                    ((uint32_t)CDIM & 0xFFFFu) << 16, // tensor_dim0 lo16
                    ((uint32_t)CDIM >> 16) | (1u << 16), // tdim0 hi | tensor_dim1=1
                    ((uint32_t)CDIM & 0xFFFFu) << 16, // tdim1 hi=0 | tile_dim0
                    1u,                               // tile_dim1=1, tile_dim2=0
                    (uint32_t)CDIM,                   // tensor_dim0_stride lo32
                    0u, 0u};                          // stride0 hi, stride1
  const u32x4 g23 = {0u, 0u, 0u, 0u};                 // dims 2..4 unused

  // Prime the pipeline: rows 0..DEPTH-1 (~1 us of compute ahead of use,
  // covering HBM latency for this WG's serial recurrence).
  if (w0)
    for (int r = 0; r < DEPTH && r < T; ++r)
      tdm_load_row(sb + (uint32_t)r * ROWBYTES, base + (size_t)r * N * CDIM,
                   g1, g23);

  float av = 0.0f;                                    // this thread's trellis alpha
  for (int t = 0; t < T; ++t) {
    // Oldest outstanding row is t. Full pipe: DEPTH rows in flight; wave 0
    // waiting down to DEPTH-1 retires exactly row t (TENSORcnt in-order).
    if (w0) {
      if (t + DEPTH <= T) __builtin_amdgcn_s_wait_tensorcnt(DEPTH - 1);
      else                __builtin_amdgcn_s_wait_tensorcnt(0);
    }
    __syncthreads();  // wave 0's retirement published to all waves

    const float* row = &sbuf[t % DEPTH][0];
    const int    cur = t & 1;

    // ---- full trellis: state tid, 5 parents, logsumexp ----
    {
      const float* ain = &abuf[cur][0];
      const int p = tid >> 2;
      float v0 = row[tid * 5 + 0] + ain[tid];
      float v1 = row[tid * 5 + 1] + ain[p];
      float v2 = row[tid * 5 + 2] + ain[p + 256];
      float v3 = row[tid * 5 + 3] + ain[p + 512];
      float v4 = row[tid * 5 + 4] + ain[p + 768];
      float mx = fmaxf(fmaxf(fmaxf(v0, v1), fmaxf(v2, v3)), v4);
      float sm = __expf(v0 - mx) + __expf(v1 - mx) + __expf(v2 - mx) +
                 __expf(v3 - mx) + __expf(v4 - mx);
      av = mx + __logf(sm);
      abuf[cur ^ 1][tid] = av;
    }

    // ---- banded CTC: state tid < nst, stay + move-in, logaddexp ----
    if (tid < nst) {
      float x = bbuf[cur][tid] + row[stay_s[tid]];
      float y = (tid > 0) ? (bbuf[cur][tid - 1] + row[move_s[tid]]) : NEGINF;
      float mx = fmaxf(x, y);
      float mn = fminf(x, y);
      bbuf[cur ^ 1][tid] = mx + __logf(1.0f + __expf(mn - mx));
    }
    __syncthreads();  // all reads of sbuf[t%DEPTH] done before TDM re-fills it

    if (w0 && (t + DEPTH < T))
      tdm_load_row(sb + (uint32_t)((t + DEPTH) % DEPTH) * ROWBYTES,
                   base + (size_t)(t + DEPTH) * N * CDIM, g1, g23);
  }

  // ---- block logsumexp over 1024 trellis alphas -> logZ_full ----
  red[tid] = av;
  __syncthreads();
  for (int off = 512; off > 0; off >>= 1) {
    if (tid < off) red[tid] = fmaxf(red[tid], red[tid + off]);
    __syncthreads();
  }
  const float mx = red[0];
  __syncthreads();
  red[tid] = __expf(av - mx);
  __syncthreads();
  for (int off = 512; off > 0; off >>= 1) {
    if (tid < off) red[tid] += red[tid + off];
    __syncthreads();
  }

  if (tid == 0) {
    float logz_full = mx + __logf(red[0]);
    int   tl        = tlen[n];
    // lengths = tl + 1 - 5; final index = lengths - 1 = tl - 5
    float logz_raw  = bbuf[T & 1][tl - 5];
    float val = -((logz_raw - logz_full) / (float)tl) / (float)N;
    atomicAdd(out, val);
  }
}

extern "C" void kernel_launch(void* const* d_in, const int* in_sizes, int n_in,
                              void* d_out, int out_size, void* d_ws, size_t ws_size,
                              hipStream_t stream) {
  (void)n_in; (void)out_size; (void)d_ws; (void)ws_size;
  const float* scores  = (const float*)d_in[0];
  const int*   targets = (const int*)d_in[1];
  const int*   tlens   = (const int*)d_in[2];
  float*       out     = (float*)d_out;

  const int N = in_sizes[2];                 // 32
  const int L = in_sizes[1] / N;             // 400
  const int T = in_sizes[0] / (N * CDIM);    // 500

  zero_out_kernel<<<1, 1, 0, stream>>>(out);
  crf_loss_fused_kernel<<<N, 1024, 0, stream>>>(scores, targets, tlens, out, T, N, L);
}